// PointMatching_42666205119331
// MI455X (gfx1250) — compile-verified
//
#include <hip/hip_runtime.h>
#include <math.h>

// ---------------- constants (match reference) ----------------
#define PP 1024
#define NN 128
#define KTOP 3
#define THRESH 0.05f
#define RAD2 0.01f          // RADIUS^2
#define MIN_CORRN 3
#define MAX_CORRN 1500
#define N_ITERN 5
#define EPSV 1e-5f
#define MAXCAND (PP * NN * KTOP) /* 393216: hard bound on corr pairs */
#define NTILES ((MAX_CORRN + 15) / 16) /* 94 */

// d_out layout: ref_pts[1500*3] | src_pts[1500*3] | corr_scores[1500] | T[16]
#define OFF_REF 0
#define OFF_SRC 4500
#define OFF_SCORE 9000
#define OFF_T 10500

typedef float v2f __attribute__((ext_vector_type(2)));
typedef float v8f __attribute__((ext_vector_type(8)));

// ---------------- small dense algebra helpers ----------------
__device__ inline float det3f(const float* M) {
  return M[0] * (M[4] * M[8] - M[5] * M[7]) - M[1] * (M[3] * M[8] - M[5] * M[6]) +
         M[2] * (M[3] * M[7] - M[4] * M[6]);
}

// Kabsch from 3x3 H (= src'^T W ref'), centroids sc/rc; writes 4x4 row-major T.
// SVD via cyclic Jacobi on H^T H; R = V diag(1,1,det(V U^T)) U^T ; t = rc - R sc.
__device__ void kabsch3(const float* Hin, const float* sc, const float* rc, bool bad,
                        float* T) {
  float Hm[9];
  if (bad) {  // reference's full-rank guard diag(3,2,1)
    Hm[0] = 3.f; Hm[1] = 0.f; Hm[2] = 0.f;
    Hm[3] = 0.f; Hm[4] = 2.f; Hm[5] = 0.f;
    Hm[6] = 0.f; Hm[7] = 0.f; Hm[8] = 1.f;
  } else {
    for (int q = 0; q < 9; ++q) Hm[q] = Hin[q];
  }
  // A = H^T H
  float A[9];
  for (int a = 0; a < 3; ++a)
    for (int b = 0; b < 3; ++b)
      A[a * 3 + b] = Hm[a] * Hm[b] + Hm[3 + a] * Hm[3 + b] + Hm[6 + a] * Hm[6 + b];
  float V[9] = {1.f, 0.f, 0.f, 0.f, 1.f, 0.f, 0.f, 0.f, 1.f};
  for (int sw = 0; sw < 15; ++sw) {
    for (int pr = 0; pr < 3; ++pr) {
      int pp = (pr == 2) ? 1 : 0;
      int qq = (pr == 0) ? 1 : 2;
      float apq = A[pp * 3 + qq];
      if (fabsf(apq) < 1e-32f) continue;
      float tau = (A[qq * 3 + qq] - A[pp * 3 + pp]) / (2.0f * apq);
      float tt = ((tau >= 0.0f) ? 1.0f : -1.0f) / (fabsf(tau) + sqrtf(1.0f + tau * tau));
      float cc = 1.0f / sqrtf(1.0f + tt * tt), ss = tt * cc;
      for (int k = 0; k < 3; ++k) {
        float x = A[k * 3 + pp], y = A[k * 3 + qq];
        A[k * 3 + pp] = cc * x - ss * y;
        A[k * 3 + qq] = ss * x + cc * y;
      }
      for (int k = 0; k < 3; ++k) {
        float x = A[pp * 3 + k], y = A[qq * 3 + k];
        A[pp * 3 + k] = cc * x - ss * y;
        A[qq * 3 + k] = ss * x + cc * y;
      }
      for (int k = 0; k < 3; ++k) {
        float x = V[k * 3 + pp], y = V[k * 3 + qq];
        V[k * 3 + pp] = cc * x - ss * y;
        V[k * 3 + qq] = ss * x + cc * y;
      }
    }
  }
  float lam[3] = {A[0], A[4], A[8]};
  for (int a = 0; a < 2; ++a)
    for (int b = a + 1; b < 3; ++b)
      if (lam[b] > lam[a]) {
        float tl = lam[a]; lam[a] = lam[b]; lam[b] = tl;
        for (int k = 0; k < 3; ++k) {
          float tv = V[k * 3 + a]; V[k * 3 + a] = V[k * 3 + b]; V[k * 3 + b] = tv;
        }
      }
  float U[9];
  for (int k = 0; k < 3; ++k) {
    float vx = V[0 + k], vy = V[3 + k], vz = V[6 + k];
    float ux = Hm[0] * vx + Hm[1] * vy + Hm[2] * vz;
    float uy = Hm[3] * vx + Hm[4] * vy + Hm[5] * vz;
    float uz = Hm[6] * vx + Hm[7] * vy + Hm[8] * vz;
    float len = sqrtf(ux * ux + uy * uy + uz * uz);
    if (len > 1e-12f) {
      U[0 + k] = ux / len; U[3 + k] = uy / len; U[6 + k] = uz / len;
    } else if (k == 0) {
      U[0] = 1.f; U[3] = 0.f; U[6] = 0.f;
    } else if (k == 1) {
      float ax = U[0], ay = U[3], az = U[6];
      float wx = ay, wy = -ax, wz = 0.f;
      float l2 = sqrtf(wx * wx + wy * wy);
      if (l2 < 1e-6f) { wx = -az; wy = 0.f; wz = ax; l2 = sqrtf(wx * wx + wz * wz); }
      U[1] = wx / l2; U[4] = wy / l2; U[7] = wz / l2;
    } else {
      float ax = U[0], ay = U[3], az = U[6], bx = U[1], by = U[4], bz = U[7];
      U[2] = ay * bz - az * by; U[5] = az * bx - ax * bz; U[8] = ax * by - ay * bx;
    }
  }
  float dsig = det3f(V) * det3f(U);
  float R[9];
  for (int c = 0; c < 3; ++c)
    for (int d = 0; d < 3; ++d)
      R[c * 3 + d] = V[c * 3 + 0] * U[d * 3 + 0] + V[c * 3 + 1] * U[d * 3 + 1] +
                     dsig * V[c * 3 + 2] * U[d * 3 + 2];
  float tx = rc[0] - (R[0] * sc[0] + R[1] * sc[1] + R[2] * sc[2]);
  float ty = rc[1] - (R[3] * sc[0] + R[4] * sc[1] + R[5] * sc[2]);
  float tz = rc[2] - (R[6] * sc[0] + R[7] * sc[1] + R[8] * sc[2]);
  T[0] = R[0]; T[1] = R[1]; T[2] = R[2]; T[3] = tx;
  T[4] = R[3]; T[5] = R[4]; T[6] = R[5]; T[7] = ty;
  T[8] = R[6]; T[9] = R[7]; T[10] = R[8]; T[11] = tz;
  T[12] = 0.f; T[13] = 0.f; T[14] = 0.f; T[15] = 1.f;
}

// S[16]: M(9) raw moments sum w*src_c*ref_d, S[9..11]=sum w*src, S[12..14]=sum w*ref, S[15]=sum w
__device__ void transform_from_sums(const float* S, float* T) {
  float tot = S[15];
  float inv = 1.0f / (tot + EPSV);
  float sc[3] = {S[9] * inv, S[10] * inv, S[11] * inv};
  float rc[3] = {S[12] * inv, S[13] * inv, S[14] * inv};
  float H[9];
  for (int c = 0; c < 3; ++c)
    for (int d = 0; d < 3; ++d)
      H[c * 3 + d] =
          (S[c * 3 + d] - sc[c] * S[12 + d] - rc[d] * S[9 + c] + sc[c] * rc[d] * tot) * inv;
  kabsch3(H, sc, rc, tot < EPSV, T);
}

// ---------------- K0: init ----------------
__global__ void pm_init(int* counter) {
  if (threadIdx.x == 0) *counter = 0;
}

// ---------------- K1: per-proposal exp + top3 + corr gather + moments ----------------
__global__ void __launch_bounds__(256) pm_score(
    const float* __restrict__ refp, const float* __restrict__ srcp,
    const unsigned char* __restrict__ refm, const unsigned char* __restrict__ srcm,
    const float* __restrict__ smap, float* __restrict__ propH, int* __restrict__ propCnt,
    float* __restrict__ candS, int* __restrict__ candF, int* __restrict__ counter) {
  __shared__ float rowv[NN][KTOP];
  __shared__ int rowi[NN][KTOP];
  __shared__ int coli[NN][KTOP];
  __shared__ unsigned char rM[NN], sM[NN];
  __shared__ float part[17][NN];
  __shared__ float red[17];
  int p = blockIdx.x;
  int tid = threadIdx.x;
  const float* base = smap + (size_t)p * NN * NN;
  if (tid < NN) { rM[tid] = refm[p * NN + tid]; sM[tid] = srcm[p * NN + tid]; }
  if (tid < NN) {  // row top-3 over src dim (stable ties: lower index first)
    int i = tid;
    const float* row = base + i * NN;
    __builtin_prefetch(row, 0, 0);
    float v0 = -1e30f, v1 = -1e30f, v2 = -1e30f;
    int j0 = 0, j1 = 0, j2 = 0;
    for (int j = 0; j < NN; ++j) {
      float e = expf(row[j]);
      if (e > v0) { v2 = v1; j2 = j1; v1 = v0; j1 = j0; v0 = e; j0 = j; }
      else if (e > v1) { v2 = v1; j2 = j1; v1 = e; j1 = j; }
      else if (e > v2) { v2 = e; j2 = j; }
    }
    rowv[i][0] = v0; rowv[i][1] = v1; rowv[i][2] = v2;
    rowi[i][0] = j0; rowi[i][1] = j1; rowi[i][2] = j2;
  } else {  // column top-3 over ref dim
    int j = tid - NN;
    float v0 = -1e30f, v1 = -1e30f, v2 = -1e30f;
    int i0 = 0, i1 = 0, i2 = 0;
    for (int i = 0; i < NN; ++i) {
      float e = expf(base[i * NN + j]);
      if (e > v0) { v2 = v1; i2 = i1; v1 = v0; i1 = i0; v0 = e; i0 = i; }
      else if (e > v1) { v2 = v1; i2 = i1; v1 = e; i1 = i; }
      else if (e > v2) { v2 = e; i2 = i; }
    }
    coli[j][0] = i0; coli[j][1] = i1; coli[j][2] = i2;
  }
  __syncthreads();
  if (tid < NN) {
    float acc[17];
    for (int q = 0; q < 17; ++q) acc[q] = 0.0f;
    int i = tid;
    bool rmv = rM[i] != 0;
    for (int t = 0; t < KTOP; ++t) {
      int j = rowi[i][t];
      float v = rowv[i][t];  // = exp score at (i,j) = score_map value at a corr
      if (v <= THRESH) continue;
      if (!rmv || sM[j] == 0) continue;
      if (coli[j][0] != i && coli[j][1] != i && coli[j][2] != i) continue;
      const float* sp = srcp + ((size_t)p * NN + j) * 3;
      const float* rp = refp + ((size_t)p * NN + i) * 3;
      float sx = sp[0], sy = sp[1], sz = sp[2];
      float rx = rp[0], ry = rp[1], rz = rp[2];
      acc[0] += v * sx * rx; acc[1] += v * sx * ry; acc[2] += v * sx * rz;
      acc[3] += v * sy * rx; acc[4] += v * sy * ry; acc[5] += v * sy * rz;
      acc[6] += v * sz * rx; acc[7] += v * sz * ry; acc[8] += v * sz * rz;
      acc[9] += v * sx; acc[10] += v * sy; acc[11] += v * sz;
      acc[12] += v * rx; acc[13] += v * ry; acc[14] += v * rz;
      acc[15] += v; acc[16] += 1.0f;
      int pos = atomicAdd(counter, 1);
      if (pos < MAXCAND) { candS[pos] = v; candF[pos] = (p * NN + i) * NN + j; }
    }
    for (int q = 0; q < 17; ++q) part[q][i] = acc[q];
  }
  __syncthreads();
  if (tid < 17) {  // fixed-order reduction => deterministic fp
    float s = 0.0f;
    for (int k = 0; k < NN; ++k) s += part[tid][k];
    red[tid] = s;
  }
  __syncthreads();
  if (tid == 0) {
    float tot = red[15];
    float inv = 1.0f / (tot + EPSV);
    float scv[3] = {red[9] * inv, red[10] * inv, red[11] * inv};
    float rcv[3] = {red[12] * inv, red[13] * inv, red[14] * inv};
    float* Hs = propH + p * 16;
    for (int c = 0; c < 3; ++c)
      for (int d = 0; d < 3; ++d)
        Hs[c * 3 + d] = (red[c * 3 + d] - scv[c] * red[12 + d] - rcv[d] * red[9 + c] +
                         scv[c] * rcv[d] * tot) * inv;
    Hs[9] = scv[0]; Hs[10] = scv[1]; Hs[11] = scv[2];
    Hs[12] = rcv[0]; Hs[13] = rcv[1]; Hs[14] = rcv[2];
    Hs[15] = tot;
    propCnt[p] = (int)(red[16] + 0.5f);
  }
}

// ---------------- K2: per-proposal Kabsch ----------------
__global__ void __launch_bounds__(256) pm_kabsch_all(const float* __restrict__ propH,
                                                     float* __restrict__ trans) {
  int p = blockIdx.x * blockDim.x + threadIdx.x;
  if (p >= PP) return;
  const float* Hs = propH + p * 16;
  float H[9];
  for (int q = 0; q < 9; ++q) H[q] = Hs[q];
  float sc[3] = {Hs[9], Hs[10], Hs[11]}, rc[3] = {Hs[12], Hs[13], Hs[14]};
  float T[16];
  kabsch3(H, sc, rc, Hs[15] < EPSV, T);
  float* o = trans + p * 16;
  for (int q = 0; q < 16; ++q) o[q] = T[q];
}

// ---------------- K3: deterministic rank-based top-1500 selection ----------------
__global__ void __launch_bounds__(256) pm_select(
    const float* __restrict__ candS, const int* __restrict__ candF,
    const int* __restrict__ counter, const float* __restrict__ refp,
    const float* __restrict__ srcp, float* __restrict__ out) {
  __shared__ float sS[256];
  __shared__ int sF[256];
  int M = *counter;
  if (M > MAXCAND) M = MAXCAND;
  int tid = threadIdx.x;
  int r = blockIdx.x * 256 + tid;
  if (r >= M && r < MAX_CORRN) {  // padded slots (only when fewer than 1500 corrs)
    int f = r - M;
    int pi = f / (NN * NN), rem = f % (NN * NN), ii = rem / NN, jj = rem % NN;
    out[OFF_SCORE + r] = 0.0f;
    for (int cc = 0; cc < 3; ++cc) {
      out[OFF_REF + r * 3 + cc] = refp[((size_t)pi * NN + ii) * 3 + cc];
      out[OFF_SRC + r * 3 + cc] = srcp[((size_t)pi * NN + jj) * 3 + cc];
    }
  }
  if ((int)(blockIdx.x * 256) >= M) return;  // uniform early-out per block
  bool act = r < M;
  float ms = 0.0f;
  int mf = 0;
  if (act) { ms = candS[r]; mf = candF[r]; }
  int rank = 0;  // total order: score desc, flat index asc (matches lax.top_k ties)
  for (int b2 = 0; b2 < M; b2 += 256) {
    int k = b2 + tid;
    sS[tid] = (k < M) ? candS[k] : -1e30f;
    sF[tid] = (k < M) ? candF[k] : 0x7fffffff;
    __syncthreads();
    int lim = M - b2;
    if (lim > 256) lim = 256;
    if (act)
      for (int q = 0; q < lim; ++q) {
        float s = sS[q];
        int f = sF[q];
        rank += ((s > ms) || (s == ms && f < mf)) ? 1 : 0;
      }
    __syncthreads();
  }
  if (act && rank < MAX_CORRN) {
    int pi = mf / (NN * NN), rem = mf % (NN * NN), ii = rem / NN, jj = rem % NN;
    out[OFF_SCORE + rank] = ms;
    for (int cc = 0; cc < 3; ++cc) {
      out[OFF_REF + rank * 3 + cc] = refp[((size_t)pi * NN + ii) * 3 + cc];
      out[OFF_SRC + rank * 3 + cc] = srcp[((size_t)pi * NN + jj) * 3 + cc];
    }
  }
}

// ---------------- K4: WMMA inlier counting (1024 proposals x 1500 pts) ----------------
// V_WMMA_F32_16X16X4_F32: A = 16 points x (x,y,z,1) homogeneous, B = [R|t]^T padded
// to 4x16 (cols>=3 zero). Exact f32 math; K=4 matches the homogeneous transform.
// All loads are unconditional with clamped indices (branchless loop body so the
// compiler can clause the loads and keep EXEC untouched around the WMMA).
__global__ void __launch_bounds__(128) pm_inliers(const float* __restrict__ trans,
                                                  const int* __restrict__ cnts,
                                                  const float* __restrict__ out,
                                                  int* __restrict__ icount) {
  int p = blockIdx.x;
  const float* T = trans + p * 16;
  float R[9] = {T[0], T[1], T[2], T[4], T[5], T[6], T[8], T[9], T[10]};
  float t3[3] = {T[3], T[7], T[11]};
  int tid = threadIdx.x;
  int lane = tid & 31, wv = tid >> 5;
  int n = lane & 15, hl = lane >> 4;
  int nc = (n < 3) ? n : 2;                // clamped column for safe loads
  float bmask = (n < 3) ? 1.0f : 0.0f;     // zero out padding columns
  // B layout (4x16, K split across lane halves): lanes 0-15 hold K=0,1; 16-31 hold K=2,3
  v2f b;
  if (hl == 0) {
    b.x = bmask * R[nc * 3 + 0];
    b.y = bmask * R[nc * 3 + 1];
  } else {
    b.x = bmask * R[nc * 3 + 2];
    b.y = bmask * t3[nc];
  }
  int cnt = 0;
  for (int tile = wv; tile < NTILES; tile += 4) {
    int m = tile * 16 + n;  // A row M = lane&15 for both halves
    int mc = (m < MAX_CORRN) ? m : (MAX_CORRN - 1);
    float scr = out[OFF_SCORE + mc];
    // padded candidates and tail rows pushed far away -> never inliers
    float off = (scr > 0.0f && m < MAX_CORRN) ? 0.0f : 1e6f;
    float sx = out[OFF_SRC + mc * 3 + 0] + off;
    float sy = out[OFF_SRC + mc * 3 + 1] + off;
    float sz = out[OFF_SRC + mc * 3 + 2] + off;
    v2f a;  // A layout: lanes 0-15 K=0,1 ; lanes 16-31 K=2,3
    if (hl == 0) { a.x = sx; a.y = sy; } else { a.x = sz; a.y = 1.0f; }
    v8f c = {0.f, 0.f, 0.f, 0.f, 0.f, 0.f, 0.f, 0.f};
    c = __builtin_amdgcn_wmma_f32_16x16x4_f32(false, a, false, b, (short)0, c, false,
                                              false);
    // D layout: lane holds column n, rows v (+8 for upper half) -> gather x,y,z via shfl
    float sq[8];
#pragma unroll
    for (int v = 0; v < 8; ++v) {
      int pm = tile * 16 + v + 8 * hl;
      int pmc = (pm < MAX_CORRN) ? pm : (MAX_CORRN - 1);
      float rv = out[OFF_REF + pmc * 3 + nc];  // unconditional clamped load
      float dd = (c[v] - rv) * bmask;          // columns >=3 contribute 0
      sq[v] = dd * dd;
    }
    int bs = hl * 16;
#pragma unroll
    for (int v = 0; v < 8; ++v) {
      float d2 = __shfl(sq[v], bs + 0, 32) + __shfl(sq[v], bs + 1, 32) +
                 __shfl(sq[v], bs + 2, 32);
      int pm = tile * 16 + v + 8 * hl;
      if (n == 0 && pm < MAX_CORRN && d2 < RAD2) cnt++;
    }
  }
  __shared__ int scnt[128];
  scnt[tid] = cnt;
  __syncthreads();
  for (int s = 64; s > 0; s >>= 1) {
    if (tid < s) scnt[tid] += scnt[tid + s];
    __syncthreads();
  }
  if (tid == 0) icount[p] = (cnts[p] >= MIN_CORRN) ? scnt[0] : -1;
}

// ---------------- K5: argmax (first max wins, like jnp.argmax) ----------------
__global__ void __launch_bounds__(256) pm_argmax(const int* __restrict__ icount,
                                                 int* __restrict__ best) {
  __shared__ int bv[256], bi[256];
  int tid = threadIdx.x;
  int mv = -2147483647, mi = 0;
  for (int k = tid; k < PP; k += 256) {
    int v = icount[k];
    if (v > mv) { mv = v; mi = k; }
  }
  bv[tid] = mv;
  bi[tid] = mi;
  __syncthreads();
  for (int s = 128; s > 0; s >>= 1) {
    if (tid < s) {
      if (bv[tid + s] > bv[tid] || (bv[tid + s] == bv[tid] && bi[tid + s] < bi[tid])) {
        bv[tid] = bv[tid + s];
        bi[tid] = bi[tid + s];
      }
    }
    __syncthreads();
  }
  if (tid == 0) *best = bi[0];
}

// ---------------- K6: 5-iteration weighted Procrustes refinement ----------------
__global__ void __launch_bounds__(256) pm_refine(const float* __restrict__ trans,
                                                 const int* __restrict__ best,
                                                 float* __restrict__ out) {
  __shared__ float part[16][256];
  __shared__ float red[16];
  __shared__ float curT[16];
  int tid = threadIdx.x;
  if (tid < 16) curT[tid] = trans[(*best) * 16 + tid];
  __syncthreads();
  for (int iter = 0; iter < N_ITERN; ++iter) {
    float acc[16];
    for (int q = 0; q < 16; ++q) acc[q] = 0.0f;
    for (int m = tid; m < MAX_CORRN; m += 256) {
      float w0 = out[OFF_SCORE + m];
      float rx = out[OFF_REF + m * 3 + 0], ry = out[OFF_REF + m * 3 + 1],
            rz = out[OFF_REF + m * 3 + 2];
      float sx = out[OFF_SRC + m * 3 + 0], sy = out[OFF_SRC + m * 3 + 1],
            sz = out[OFF_SRC + m * 3 + 2];
      float off = (w0 > 0.0f) ? 0.0f : 1e6f;  // src_eval for padded candidates
      float ex = sx + off, ey = sy + off, ez = sz + off;
      float ax = curT[0] * ex + curT[1] * ey + curT[2] * ez + curT[3];
      float ay = curT[4] * ex + curT[5] * ey + curT[6] * ez + curT[7];
      float az = curT[8] * ex + curT[9] * ey + curT[10] * ez + curT[11];
      float dx = rx - ax, dy = ry - ay, dz = rz - az;
      float d2 = dx * dx + dy * dy + dz * dz;
      float w = (d2 < RAD2) ? w0 : 0.0f;
      acc[0] += w * sx * rx; acc[1] += w * sx * ry; acc[2] += w * sx * rz;
      acc[3] += w * sy * rx; acc[4] += w * sy * ry; acc[5] += w * sy * rz;
      acc[6] += w * sz * rx; acc[7] += w * sz * ry; acc[8] += w * sz * rz;
      acc[9] += w * sx; acc[10] += w * sy; acc[11] += w * sz;
      acc[12] += w * rx; acc[13] += w * ry; acc[14] += w * rz;
      acc[15] += w;
    }
    for (int q = 0; q < 16; ++q) part[q][tid] = acc[q];
    __syncthreads();
    if (tid < 16) {
      float s = 0.0f;
      for (int k = 0; k < 256; ++k) s += part[tid][k];
      red[tid] = s;
    }
    __syncthreads();
    if (tid == 0) {
      float T[16];
      transform_from_sums(red, T);
      for (int q = 0; q < 16; ++q) curT[q] = T[q];
    }
    __syncthreads();
  }
  if (tid < 16) out[OFF_T + tid] = curT[tid];
}

// ---------------- launcher ----------------
extern "C" void kernel_launch(void* const* d_in, const int* in_sizes, int n_in,
                              void* d_out, int out_size, void* d_ws, size_t ws_size,
                              hipStream_t stream) {
  const float* refp = (const float*)d_in[0];
  const float* srcp = (const float*)d_in[1];
  const unsigned char* refm = (const unsigned char*)d_in[2];  // jax bool = 1 byte
  const unsigned char* srcm = (const unsigned char*)d_in[3];
  const float* smap = (const float*)d_in[4];
  float* out = (float*)d_out;

  char* w = (char*)d_ws;
  int* counter = (int*)w;                                  // 16 B reserved
  float* candS = (float*)(w + 16);                         // MAXCAND f32
  int* candF = (int*)(w + 16 + (size_t)MAXCAND * 4);       // MAXCAND i32
  float* propH = (float*)(w + 16 + (size_t)MAXCAND * 8);   // PP*16 f32
  int* propCnt = (int*)((char*)propH + (size_t)PP * 16 * 4);
  float* trans = (float*)((char*)propCnt + (size_t)PP * 4);  // PP*16 f32
  int* icount = (int*)((char*)trans + (size_t)PP * 16 * 4);
  int* best = (int*)((char*)icount + (size_t)PP * 4);

  pm_init<<<1, 32, 0, stream>>>(counter);
  pm_score<<<PP, 256, 0, stream>>>(refp, srcp, refm, srcm, smap, propH, propCnt, candS,
                                   candF, counter);
  pm_kabsch_all<<<PP / 256, 256, 0, stream>>>(propH, trans);
  pm_select<<<(MAXCAND + 255) / 256, 256, 0, stream>>>(candS, candF, counter, refp, srcp,
                                                       out);
  pm_inliers<<<PP, 128, 0, stream>>>(trans, propCnt, out, icount);
  pm_argmax<<<1, 256, 0, stream>>>(icount, best);
  pm_refine<<<1, 256, 0, stream>>>(trans, best, out);
  (void)in_sizes; (void)n_in; (void)out_size; (void)ws_size;
}